// ConvLRUModel_2697239462383
// MI455X (gfx1250) — compile-verified
//
#include <hip/hip_runtime.h>
#include <hip/hip_bf16.h>

// ---------------------------------------------------------------------------
// ConvLRU forward for MI455X (gfx1250, wave32, WMMA).
// fp32 model -> V_WMMA_F32_16X16X4_F32 (precision-preserving matrix path).
// Convs dominate (~155 GFLOP); all activations fit in the 192MB L2 so the
// implicit-GEMM conv reads activations straight from global (L2-hot).
// 4-way N-blocking per wave (A-fragment reuse x4), weights pre-transposed to
// [k][o] for coalesced B loads, templated kernels (no runtime branches in
// WMMA loops), global_prefetch on the weight stream.
// ---------------------------------------------------------------------------

typedef __attribute__((ext_vector_type(2))) float v2f;
typedef __attribute__((ext_vector_type(8))) float v8f;

#define EMB   64
#define HH    32
#define WW    32
#define BB    2
#define LL    64
#define FFNC  128
#define BL    (BB * LL)          // 128 images
#define IMGSZ (HH * WW)          // 1024
#define XELEMS (BL * EMB * IMGSZ) // 8388608
#define PI_F  3.14159265358979f
#define EPSF  1e-5f

__device__ __forceinline__ v8f wmma4(v2f a, v2f b, v8f c) {
  // D = A(16x4,f32) x B(4x16,f32) + C(16x16,f32)
  return __builtin_amdgcn_wmma_f32_16x16x4_f32(
      false, a, false, b, (short)0, c, false, false);
}

// ---------------------------------------------------------------------------
// Weight transpose: src[o][k] -> dst[k][o]   (one-time prep; tiny)
// ---------------------------------------------------------------------------
__global__ void transpose_w_kernel(const float* __restrict__ src,
                                   float* __restrict__ dst, int O, int K) {
  const int gid = blockIdx.x * 256 + threadIdx.x;
  if (gid < O * K) {
    const int o = gid / K;
    const int k = gid - o * K;
    dst[k * O + o] = src[gid];
  }
}

// ---------------------------------------------------------------------------
// 2-D 32x32 DFT, forward (real input -> complex).  One block per (b,l,c) image.
// ---------------------------------------------------------------------------
__global__ void fft2_fwd_kernel(const float* __restrict__ x,
                                float* __restrict__ hr, float* __restrict__ hi) {
  __shared__ float twc[1024], tws[1024];
  __shared__ float br_[1024];
  __shared__ float mr[1024], mi[1024];
  const long base = (long)blockIdx.x * IMGSZ;
  const int tid = threadIdx.x;
  for (int idx = tid; idx < 1024; idx += 256) {
    int k = idx >> 5, w = idx & 31;
    float ang = -2.f * PI_F * (float)((k * w) & 31) / 32.f;
    twc[idx] = cosf(ang);
    tws[idx] = sinf(ang);
    br_[idx] = x[base + idx];
  }
  __syncthreads();
  for (int idx = tid; idx < 1024; idx += 256) {   // row DFT over w
    int u = idx >> 5, k = idx & 31;
    float sr = 0.f, si = 0.f;
    for (int w = 0; w < 32; ++w) {
      float v = br_[u * 32 + w];
      sr += v * twc[k * 32 + w];
      si += v * tws[k * 32 + w];
    }
    mr[idx] = sr; mi[idx] = si;
  }
  __syncthreads();
  for (int idx = tid; idx < 1024; idx += 256) {   // column DFT over u
    int ky = idx >> 5, kx = idx & 31;
    float sr = 0.f, si = 0.f;
    for (int u = 0; u < 32; ++u) {
      float ar = mr[u * 32 + kx], ai = mi[u * 32 + kx];
      float c = twc[ky * 32 + u], s = tws[ky * 32 + u];
      sr += ar * c - ai * s;
      si += ar * s + ai * c;
    }
    hr[base + idx] = sr;
    hi[base + idx] = si;
  }
}

// 2-D 32x32 inverse DFT (complex -> complex, scaled by 1/1024)
__global__ void fft2_inv_kernel(const float* __restrict__ xr, const float* __restrict__ xi,
                                float* __restrict__ hr, float* __restrict__ hi) {
  __shared__ float twc[1024], tws[1024];
  __shared__ float br_[1024], bi_[1024];
  __shared__ float mr[1024], mi[1024];
  const long base = (long)blockIdx.x * IMGSZ;
  const int tid = threadIdx.x;
  for (int idx = tid; idx < 1024; idx += 256) {
    int k = idx >> 5, w = idx & 31;
    float ang = 2.f * PI_F * (float)((k * w) & 31) / 32.f;
    twc[idx] = cosf(ang);
    tws[idx] = sinf(ang);
    br_[idx] = xr[base + idx];
    bi_[idx] = xi[base + idx];
  }
  __syncthreads();
  for (int idx = tid; idx < 1024; idx += 256) {
    int u = idx >> 5, k = idx & 31;
    float sr = 0.f, si = 0.f;
    for (int w = 0; w < 32; ++w) {
      float ar = br_[u * 32 + w], ai = bi_[u * 32 + w];
      float c = twc[k * 32 + w], s = tws[k * 32 + w];
      sr += ar * c - ai * s;
      si += ar * s + ai * c;
    }
    mr[idx] = sr; mi[idx] = si;
  }
  __syncthreads();
  for (int idx = tid; idx < 1024; idx += 256) {
    int ky = idx >> 5, kx = idx & 31;
    float sr = 0.f, si = 0.f;
    for (int u = 0; u < 32; ++u) {
      float ar = mr[u * 32 + kx], ai = mi[u * 32 + kx];
      float c = twc[ky * 32 + u], s = tws[ky * 32 + u];
      sr += ar * c - ai * s;
      si += ar * s + ai * c;
    }
    hr[base + idx] = sr * (1.f / 1024.f);
    hi[base + idx] = si * (1.f / 1024.f);
  }
}

// ---------------------------------------------------------------------------
// Complex channel mix: out[b,l,o,hw] = sum_c in[b,l,c,hw] * W[o,c]  (+ bias)
// M = B*L*H*W rows, N = 64 (all 4 16-wide tiles per wave), K = 64.
// Weights pre-transposed to wT[c][o] for coalesced B loads.
// MODE 0: (h*W + bb) * exp(gamma_log) -> complex out (B-mix)
// MODE 1: real(h*W) + cb_r            -> real out     (C-mix, half the WMMAs)
// ---------------------------------------------------------------------------
template <int MODE>
__global__ void cmix_wmma_kernel(const float* __restrict__ ar, const float* __restrict__ ai,
                                 const float* __restrict__ wTr, const float* __restrict__ wTi,
                                 const float* __restrict__ biasr, const float* __restrict__ biasi,
                                 const float* __restrict__ gamma_log,
                                 float* __restrict__ outr, float* __restrict__ outi) {
  const int lane = threadIdx.x & 31;
  const int rowTile = blockIdx.x * 8 + (threadIdx.x >> 5);
  const int rowBase = rowTile * 16;

  // A fragment (ISA 7.12.2, 32-bit A 16x4): lanes 0-15 K0/K1, lanes 16-31 K2/K3
  const int m  = lane & 15;
  const int kh = (lane >> 4) << 1;
  const int p  = rowBase + m;                  // global row = (b*l)*1024 + hw
  const int bl = p >> 10, hw = p & 1023;
  const long abase = (long)bl * (EMB * IMGSZ) + hw;
  const int o0 = lane & 15;                    // output channel, +16 per tile

  v8f accr[4] = {};
  v8f acci[4] = {};
  for (int k0 = 0; k0 < EMB; k0 += 4) {
    const int c0 = k0 + kh;
    v2f Ar, Ai;
    Ar.x = ar[abase + (long)c0 * IMGSZ];
    Ar.y = ar[abase + (long)(c0 + 1) * IMGSZ];
    Ai.x = ai[abase + (long)c0 * IMGSZ];
    Ai.y = ai[abase + (long)(c0 + 1) * IMGSZ];
#pragma unroll
    for (int t4 = 0; t4 < 4; ++t4) {
      const int o = o0 + 16 * t4;
      v2f Br, Bi, nBi;
      Br.x = wTr[c0 * EMB + o];       Br.y = wTr[(c0 + 1) * EMB + o];
      Bi.x = wTi[c0 * EMB + o];       Bi.y = wTi[(c0 + 1) * EMB + o];
      nBi.x = -Bi.x; nBi.y = -Bi.y;
      accr[t4] = wmma4(Ar, Br, accr[t4]);      // Re*Re
      accr[t4] = wmma4(Ai, nBi, accr[t4]);     // -Im*Im
      if (MODE == 0) {
        acci[t4] = wmma4(Ar, Bi, acci[t4]);    // Re*Im
        acci[t4] = wmma4(Ai, Br, acci[t4]);    // Im*Re
      }
    }
  }
  // D layout: VGPR i -> M = i (+8 for lanes 16..31), N = lane&15
  const int mb = (lane >> 4) << 3;
#pragma unroll
  for (int t4 = 0; t4 < 4; ++t4) {
    const int o = o0 + 16 * t4;
    const float bvr = biasr[o];
    const float bvi = (MODE == 0) ? biasi[o] : 0.f;
#pragma unroll
    for (int i = 0; i < 8; ++i) {
      const int pr  = rowBase + mb + i;
      const int bl2 = pr >> 10, hw2 = pr & 1023;
      const long oaddr = (long)bl2 * (EMB * IMGSZ) + (long)o * IMGSZ + hw2;
      if (MODE == 0) {
        const float g = expf(gamma_log[o * 32 + (hw2 >> 5)]);
        outr[oaddr] = (accr[t4][i] + bvr) * g;
        outi[oaddr] = (acci[t4][i] + bvi) * g;
      } else {
        outr[oaddr] = accr[t4][i] + bvr;
      }
    }
  }
}

// ---------------------------------------------------------------------------
// LRU associative scan over L: y[l] = lamb*y[l-1] + h[l].  131072 chains.
// ---------------------------------------------------------------------------
__global__ void lru_scan_kernel(float* __restrict__ tr, float* __restrict__ ti,
                                const float* __restrict__ nu_log,
                                const float* __restrict__ theta_log,
                                float* __restrict__ lastout) {
  const int gid = blockIdx.x * blockDim.x + threadIdx.x;  // B*EMB*IMGSZ
  const int b = gid >> 16;
  const int chw = gid & 65535;
  const int c = chw >> 10;
  const int hw = chw & 1023;
  const int hrow = hw >> 5;
  const float nu = expf(nu_log[c * 32 + hrow]);
  const float th = expf(theta_log[c * 32 + hrow]);
  const float mag = expf(-nu);
  const float lr = mag * cosf(th), li = mag * sinf(th);
  float accr = 0.f, acci = 0.f;
  const long base = (long)b * ((long)LL * EMB * IMGSZ) + (long)c * IMGSZ + hw;
  for (int l = 0; l < LL; ++l) {
    const long a = base + (long)l * (EMB * IMGSZ);
    const float vr = tr[a], vi = ti[a];
    const float nr = lr * accr - li * acci + vr;
    const float ni = lr * acci + li * accr + vi;
    accr = nr; acci = ni;
    tr[a] = nr; ti[a] = ni;
  }
  const long lo = (((long)b * EMB + c) * IMGSZ + hw) * 2;
  lastout[lo]     = accr;
  lastout[lo + 1] = acci;
}

// ---------------------------------------------------------------------------
// LayerNorm over (C,H,W) per (b,l) sample + residual:  out = LN(z)*w + b + res
// ---------------------------------------------------------------------------
__global__ void ln_residual_kernel(const float* __restrict__ z, const float* __restrict__ w,
                                   const float* __restrict__ bparm,
                                   const float* __restrict__ res,
                                   float* __restrict__ out, int N) {
  __shared__ float ssum[256], ssq[256];
  const long base = (long)blockIdx.x * N;
  float s = 0.f, sq = 0.f;
  for (int i = threadIdx.x; i < N; i += 256) {
    const float v = z[base + i];
    s += v; sq += v * v;
  }
  ssum[threadIdx.x] = s; ssq[threadIdx.x] = sq;
  __syncthreads();
  for (int off = 128; off > 0; off >>= 1) {
    if (threadIdx.x < off) {
      ssum[threadIdx.x] += ssum[threadIdx.x + off];
      ssq[threadIdx.x]  += ssq[threadIdx.x + off];
    }
    __syncthreads();
  }
  const float mean = ssum[0] / (float)N;
  const float var  = ssq[0] / (float)N - mean * mean;
  const float rstd = rsqrtf(var + EPSF);
  for (int i = threadIdx.x; i < N; i += 256) {
    out[base + i] = (z[base + i] - mean) * rstd * w[i] + bparm[i] + res[base + i];
  }
}

// ---------------------------------------------------------------------------
// 3x3 SAME conv as implicit GEMM with V_WMMA_F32_16X16X4_F32.
// Per wave: 16 spatial positions x 64 output channels (4 accumulator tiles,
// A fragment reused x4).  Weights pre-transposed wT[k][o] -> coalesced B.
// ---------------------------------------------------------------------------
__device__ __forceinline__ float convA(const float* __restrict__ img,
                                       int k, int y, int x) {
  const int c  = k / 9;
  const int rs = k % 9;
  const int iy = y + rs / 3 - 1;
  const int ix = x + rs % 3 - 1;
  if ((unsigned)iy < 32u && (unsigned)ix < 32u)
    return img[c * IMGSZ + iy * 32 + ix];
  return 0.f;
}

template <int CIN, int COUT, bool LEAKY>
__global__ void conv3_wmma_kernel(const float* __restrict__ in,
                                  const float* __restrict__ wT,   // [CIN*9][COUT]
                                  const float* __restrict__ bias,
                                  float* __restrict__ out) {
  constexpr int KTOT = CIN * 9;
  constexpr int NGRP = COUT / 64;              // 64-wide output-channel groups
  const int lane = threadIdx.x & 31;
  const int waveId = blockIdx.x * 8 + (threadIdx.x >> 5);
  const int ng  = waveId % NGRP;
  const int t   = waveId / NGRP;
  const int sp  = t & 63;                      // 64 spatial tiles of 16 / image
  const int img = t >> 6;
  const int pBase = sp * 16;

  const int m  = lane & 15;
  const int kh = (lane >> 4) << 1;
  const int p  = pBase + m;
  const int y  = p >> 5, x = p & 31;
  const float* inImg = in + (long)img * CIN * IMGSZ;
  const int oBase = ng * 64 + (lane & 15);     // + 16*t4 per tile

  v8f acc[4] = {};
  for (int k0 = 0; k0 < KTOT; k0 += 4) {
    const int ka = k0 + kh;
    // prefetch weight stream ~64 k ahead (wave-uniform branch)
    if ((k0 & 63) == 0 && (k0 + 64) < KTOT)
      __builtin_prefetch(wT + (long)(k0 + 64) * COUT + oBase, 0, 3);
    v2f A;
    A.x = convA(inImg, ka,     y, x);
    A.y = convA(inImg, ka + 1, y, x);
    const float* wk0 = wT + (long)ka * COUT + oBase;
    const float* wk1 = wk0 + COUT;
#pragma unroll
    for (int t4 = 0; t4 < 4; ++t4) {
      v2f Bv;
      Bv.x = wk0[16 * t4];
      Bv.y = wk1[16 * t4];
      acc[t4] = wmma4(A, Bv, acc[t4]);
    }
  }
  const int mb = (lane >> 4) << 3;
#pragma unroll
  for (int t4 = 0; t4 < 4; ++t4) {
    const int o = oBase + 16 * t4;
    const float bval = bias[o];
    float* outImg = out + (long)img * COUT * IMGSZ + (long)o * IMGSZ;
#pragma unroll
    for (int i = 0; i < 8; ++i) {
      float v = acc[t4][i] + bval;
      if (LEAKY) v = (v >= 0.f) ? v : 0.01f * v;
      outImg[pBase + mb + i] = v;
    }
  }
}

// ---------------------------------------------------------------------------
// Host orchestration
// ---------------------------------------------------------------------------
struct BlockP {
  const float *nu_log, *theta_log, *gamma_log;
  const float *Bw_r, *Bw_i, *Bb_r, *Bb_i;
  const float *Cw_r, *Cw_i, *Cb_r, *Cb_i;
  const float *ln1_w, *ln1_b;
  const float *cin_w, *cin_b;
  const float *hid_w, *hid_b, *hid_lnw, *hid_lnb;
  const float *cout_w, *cout_b, *ln2_w, *ln2_b;
};

extern "C" void kernel_launch(void* const* d_in, const int* in_sizes, int n_in,
                              void* d_out, int out_size, void* d_ws, size_t ws_size,
                              hipStream_t stream) {
  (void)n_in; (void)out_size; (void)ws_size;
  auto F = [&](int i) -> const float* { return (const float*)d_in[i]; };

  // Detect flattening order: insertion-order (x first) vs jax sorted keys.
  const bool xFirst = (in_sizes[0] == XELEMS);
  const float* xin;
  BlockP blk[2];
  if (xFirst) {
    xin = F(0);
    for (int k = 0; k < 2; ++k) {
      int b = 1 + k * 23;
      blk[k] = { F(b+0), F(b+1), F(b+2),            // nu_log, theta_log, gamma_log
                 F(b+3), F(b+4), F(b+5), F(b+6),    // Bw_r, Bw_i, Bb_r, Bb_i
                 F(b+7), F(b+8), F(b+9), F(b+10),   // Cw_r, Cw_i, Cb_r, Cb_i
                 F(b+11), F(b+12),                  // ln1_w, ln1_b
                 F(b+13), F(b+14),                  // cin_w, cin_b
                 F(b+15), F(b+16), F(b+17), F(b+18),// hid: w, b, ln_w, ln_b
                 F(b+19), F(b+20), F(b+21), F(b+22) // cout_w, cout_b, ln2_w, ln2_b
               };
    }
  } else {
    // sorted keys: Bb_i Bb_r Bw_i Bw_r Cb_i Cb_r Cw_i Cw_r cin_b cin_w cout_b
    //              cout_w gamma_log hid{b,ln_b,ln_w,w} ln1_b ln1_w ln2_b ln2_w
    //              nu_log theta_log ; x last.
    xin = F(46);
    for (int k = 0; k < 2; ++k) {
      int b = k * 23;
      blk[k].Bb_i = F(b+0);  blk[k].Bb_r = F(b+1);
      blk[k].Bw_i = F(b+2);  blk[k].Bw_r = F(b+3);
      blk[k].Cb_i = F(b+4);  blk[k].Cb_r = F(b+5);
      blk[k].Cw_i = F(b+6);  blk[k].Cw_r = F(b+7);
      blk[k].cin_b = F(b+8); blk[k].cin_w = F(b+9);
      blk[k].cout_b = F(b+10); blk[k].cout_w = F(b+11);
      blk[k].gamma_log = F(b+12);
      blk[k].hid_b = F(b+13); blk[k].hid_lnb = F(b+14);
      blk[k].hid_lnw = F(b+15); blk[k].hid_w = F(b+16);
      blk[k].ln1_b = F(b+17); blk[k].ln1_w = F(b+18);
      blk[k].ln2_b = F(b+19); blk[k].ln2_w = F(b+20);
      blk[k].nu_log = F(b+21); blk[k].theta_log = F(b+22);
    }
  }

  // Workspace layout (fp32), ~226MB; y2 aliases freed hr/hi.
  float* ws   = (float*)d_ws;
  float* xcur = ws;                 // 8388608
  float* hr   = xcur + XELEMS;      // 8388608
  float* hi   = hr   + XELEMS;      // 8388608
  float* tr   = hi   + XELEMS;      // 8388608
  float* ti   = tr   + XELEMS;      // 8388608
  float* yb   = ti   + XELEMS;      // 16777216 (FFN activations)
  float* y2   = hr;                 // aliases hr+hi (16777216)
  float* wpool = yb + 16777216;     // transposed-weight pool

  // per-block transposed weights
  float* cinT[2];  float* hidT[2];  float* coutT[2];
  float* BwTr[2];  float* BwTi[2];  float* CwTr[2];  float* CwTi[2];
  {
    float* q = wpool;
    for (int k = 0; k < 2; ++k) {
      cinT[k]  = q; q += FFNC * EMB * 9;   // 73728
      hidT[k]  = q; q += FFNC * FFNC * 9;  // 147456
      coutT[k] = q; q += EMB * FFNC * 9;   // 73728
      BwTr[k]  = q; q += EMB * EMB;        // 4096
      BwTi[k]  = q; q += EMB * EMB;
      CwTr[k]  = q; q += EMB * EMB;
      CwTi[k]  = q; q += EMB * EMB;
    }
  }

  float* outx = (float*)d_out;
  const dim3 blk256(256);

  // One-time weight transposes (tiny)
  for (int k = 0; k < 2; ++k) {
    const BlockP& P = blk[k];
    transpose_w_kernel<<<(FFNC*EMB*9  + 255)/256, blk256, 0, stream>>>(P.cin_w,  cinT[k],  FFNC, EMB*9);
    transpose_w_kernel<<<(FFNC*FFNC*9 + 255)/256, blk256, 0, stream>>>(P.hid_w,  hidT[k],  FFNC, FFNC*9);
    transpose_w_kernel<<<(EMB*FFNC*9  + 255)/256, blk256, 0, stream>>>(P.cout_w, coutT[k], EMB,  FFNC*9);
    transpose_w_kernel<<<(EMB*EMB + 255)/256, blk256, 0, stream>>>(P.Bw_r, BwTr[k], EMB, EMB);
    transpose_w_kernel<<<(EMB*EMB + 255)/256, blk256, 0, stream>>>(P.Bw_i, BwTi[k], EMB, EMB);
    transpose_w_kernel<<<(EMB*EMB + 255)/256, blk256, 0, stream>>>(P.Cw_r, CwTr[k], EMB, EMB);
    transpose_w_kernel<<<(EMB*EMB + 255)/256, blk256, 0, stream>>>(P.Cw_i, CwTi[k], EMB, EMB);
  }

  (void)hipMemcpyAsync(xcur, xin, (size_t)XELEMS * sizeof(float),
                       hipMemcpyDeviceToDevice, stream);

  for (int k = 0; k < 2; ++k) {
    const BlockP& P = blk[k];
    float* lastout = outx + XELEMS + (long)k * (BB * EMB * IMGSZ * 2);

    // ---- LRU layer ----
    fft2_fwd_kernel<<<BL * EMB, blk256, 0, stream>>>(xcur, hr, hi);
    // B-mix: 8192 row tiles / 8 waves = 1024 blocks; each wave does N=64
    cmix_wmma_kernel<0><<<1024, blk256, 0, stream>>>(
        hr, hi, BwTr[k], BwTi[k], P.Bb_r, P.Bb_i, P.gamma_log, tr, ti);
    lru_scan_kernel<<<512, blk256, 0, stream>>>(tr, ti, P.nu_log, P.theta_log, lastout);
    fft2_inv_kernel<<<BL * EMB, blk256, 0, stream>>>(tr, ti, hr, hi);
    // C-mix: real part only
    cmix_wmma_kernel<1><<<1024, blk256, 0, stream>>>(
        hr, hi, CwTr[k], CwTi[k], P.Cb_r, P.Cb_i, P.gamma_log, tr, ti);
    ln_residual_kernel<<<BL, blk256, 0, stream>>>(tr, P.ln1_w, P.ln1_b, xcur, xcur,
                                                  EMB * IMGSZ);

    // ---- FFN ----
    // waves = 128 img * 64 sp-tiles * (COUT/64) ; 8 waves per block
    conv3_wmma_kernel<EMB, FFNC, true><<<2048, blk256, 0, stream>>>(
        xcur, cinT[k], P.cin_b, yb);
    conv3_wmma_kernel<FFNC, FFNC, true><<<2048, blk256, 0, stream>>>(
        yb, hidT[k], P.hid_b, y2);
    ln_residual_kernel<<<BL, blk256, 0, stream>>>(y2, P.hid_lnw, P.hid_lnb, yb, yb,
                                                  FFNC * IMGSZ);
    conv3_wmma_kernel<FFNC, EMB, false><<<1024, blk256, 0, stream>>>(
        yb, coutT[k], P.cout_b, tr);
    ln_residual_kernel<<<BL, blk256, 0, stream>>>(tr, P.ln2_w, P.ln2_b, xcur, xcur,
                                                  EMB * IMGSZ);
  }

  (void)hipMemcpyAsync(outx, xcur, (size_t)XELEMS * sizeof(float),
                       hipMemcpyDeviceToDevice, stream);
}